// Attention_57346403336437
// MI455X (gfx1250) — compile-verified
//
#include <hip/hip_runtime.h>

// Reference:  a = softmax_T( relu( tanh(x@W1x + y@W1y + b1) @ W2 + b2 ) ) with
// exp/(sum+eps) normalization.  x_length (d_in[2]) is unused by the reference.
// Memory-bound: 134 MB of x dominates -> fuse everything, one block per n.

typedef float v2f __attribute__((ext_vector_type(2)));
typedef float v8f __attribute__((ext_vector_type(8)));

#define N_   256
#define T_   512
#define D_   256
#define HID_ 32
#define EPS_ 1e-7f

__global__ __launch_bounds__(256) void attn_pool_kernel(
    const float* __restrict__ x,   // (N,T,D)
    const float* __restrict__ y,   // (N,D)
    const float* __restrict__ W1,  // (2D,HID) row-major; rows [0,D)=W1x, [D,2D)=W1y
    const float* __restrict__ b1,  // (HID)
    const float* __restrict__ W2,  // (HID,1)
    const float* __restrict__ b2,  // (1)
    float* __restrict__ out)       // (N,T)
{
    __shared__ float sW1x[D_ * HID_];  // 32 KB: B-matrix operand, reused by all 8 waves
    __shared__ float sYpart[256];
    __shared__ float sYb[HID_];        // y@W1y + b1 for this n
    __shared__ float sS[T_];           // exp(relu(...)) scores
    __shared__ float sRed[8];

    const int n     = blockIdx.x;
    const int tid   = threadIdx.x;
    const int wave  = tid >> 5;        // 8 waves (wave32)
    const int lane  = tid & 31;
    const int lhalf = lane >> 4;       // 0: K={0,1}, 1: K={2,3} (A/B 16x4 f32 layout)
    const int lmod  = lane & 15;

    // ---- Stage W1x into LDS (coalesced) ----
    for (int i = tid; i < D_ * HID_; i += 256) sW1x[i] = W1[i];

    // ---- y @ W1y + b1 (tiny: D*HID MACs, split across 8 d-chunks x 32 h) ----
    {
        const int h = tid & 31, chunk = tid >> 5;
        const float* yn = y + (size_t)n * D_;
        float p = 0.f;
        const int d0 = chunk * 32;
        for (int d = d0; d < d0 + 32; ++d)
            p += yn[d] * W1[(size_t)(D_ + d) * HID_ + h];
        sYpart[tid] = p;
    }
    __syncthreads();
    if (tid < HID_) {
        float acc = b1[tid];
        for (int c = 0; c < 8; ++c) acc += sYpart[c * 32 + tid];
        sYb[tid] = acc;
    }
    __syncthreads();

    const float w2a = W2[lmod];        // col-tile 0: h = lmod
    const float w2b = W2[16 + lmod];   // col-tile 1: h = 16 + lmod
    const float b2v = b2[0];
    const float yb0 = sYb[lmod];
    const float yb1 = sYb[16 + lmod];

    // ---- main: each wave owns 16-row tiles of T; C = x_tile @ W1x via f32 WMMA ----
    for (int rt = wave; rt < T_ / 16; rt += 8) {          // uniform per wave: EXEC stays full
        const float* xrow = x + ((size_t)n * T_ + (size_t)rt * 16 + lmod) * D_;
        v8f c0 = {0.f,0.f,0.f,0.f,0.f,0.f,0.f,0.f};
        v8f c1 = {0.f,0.f,0.f,0.f,0.f,0.f,0.f,0.f};

        for (int k = 0; k < D_; k += 4) {
            // A 16x4 f32: lane holds row lmod, K = k + 2*lhalf + {0,1}  (8B-aligned load)
            v2f a = *(const v2f*)(xrow + k + 2 * lhalf);
            // B 4x16 f32 from LDS: VGPR j, lane L -> K = k + 2*lhalf + j, N = lmod
            const float* wp = &sW1x[(k + 2 * lhalf) * HID_];
            v2f bb0; bb0.x = wp[lmod];      bb0.y = wp[HID_ + lmod];
            v2f bb1; bb1.x = wp[16 + lmod]; bb1.y = wp[HID_ + 16 + lmod];
            c0 = __builtin_amdgcn_wmma_f32_16x16x4_f32(false, a, false, bb0,
                                                       (short)0, c0, false, false);
            c1 = __builtin_amdgcn_wmma_f32_16x16x4_f32(false, a, false, bb1,
                                                       (short)0, c1, false, false);
        }

        // Epilogue: C VGPR v, lane L -> row m = v + 8*lhalf, col h = (tile)*16 + lmod
        #pragma unroll
        for (int v = 0; v < 8; ++v) {
            float t0 = tanhf(c0[v] + yb0);
            float t1 = tanhf(c1[v] + yb1);
            float s  = t0 * w2a + t1 * w2b;   // partial of h1@W2 for row m
            // reduce over the 16 lanes of this half (masks < 16 stay in-half, wave32)
            s += __shfl_xor(s, 1, 32);
            s += __shfl_xor(s, 2, 32);
            s += __shfl_xor(s, 4, 32);
            s += __shfl_xor(s, 8, 32);
            if (lmod == 0) {
                float h2 = fmaxf(s + b2v, 0.f);
                sS[rt * 16 + v + 8 * lhalf] = expf(h2);
            }
        }
    }
    __syncthreads();

    // ---- normalize: a = exp / (sum_T + eps) ----
    float acc = sS[tid] + sS[tid + 256];
    acc += __shfl_xor(acc, 1, 32);
    acc += __shfl_xor(acc, 2, 32);
    acc += __shfl_xor(acc, 4, 32);
    acc += __shfl_xor(acc, 8, 32);
    acc += __shfl_xor(acc, 16, 32);
    if (lane == 0) sRed[wave] = acc;
    __syncthreads();
    if (tid == 0) {
        float tot = EPS_;
        for (int i = 0; i < 8; ++i) tot += sRed[i];
        sRed[0] = 1.0f / tot;
    }
    __syncthreads();
    const float inv = sRed[0];
    float* o = out + (size_t)n * T_;
    o[tid]       = sS[tid] * inv;
    o[tid + 256] = sS[tid + 256] * inv;
}

extern "C" void kernel_launch(void* const* d_in, const int* in_sizes, int n_in,
                              void* d_out, int out_size, void* d_ws, size_t ws_size,
                              hipStream_t stream) {
    const float* x  = (const float*)d_in[0];
    const float* y  = (const float*)d_in[1];
    // d_in[2] = x_length (int32) — unused by the reference computation
    const float* W1 = (const float*)d_in[3];
    const float* b1 = (const float*)d_in[4];
    const float* W2 = (const float*)d_in[5];
    const float* b2 = (const float*)d_in[6];
    attn_pool_kernel<<<N_, 256, 0, stream>>>(x, y, W1, b1, W2, b2, (float*)d_out);
}